// DPAG_21766894256691
// MI455X (gfx1250) — compile-verified
//
#include <hip/hip_runtime.h>

// ---------------------------------------------------------------------------
// B=8, N_SMI=100, N_ATOM=45, N_PROT=1000, D=64, Nd=145.
// Fused plan (mean commutes through the shared linear layer, killing the
// (8,145,1000,64) intermediate and its 9.5 GFLOP GEMM):
//   1) tmpSmi = smi_tf @ w_att + b ; P = pro_tf @ w_att + b     [WMMA f32]
//   2) S = concat(tmpSmi, drug_gat)
//   3) Msmi[b,i,d] = mean_j relu(S+P);  Mpro[b,j,d] = mean_i relu(S+P)
//      (register-blocked: 5 rows / 8 rows per block for L2-traffic reuse)
//   4) Gsmi = sigmoid(Msmi @ w_att + b); Gpro = sigmoid(Mpro @ w_att + b)
//   5) X[b] = [ mean_i S*(0.5+Gsmi) , mean_j pro_tf*(0.5+Gpro) ]  (16x128)
//   6) MLP 128->1024->1024->512 via WMMA, 512->2 scalar.
// ---------------------------------------------------------------------------

typedef __attribute__((ext_vector_type(2))) float v2f;
typedef __attribute__((ext_vector_type(8))) float v8f;

#define ACT_NONE 0
#define ACT_RELU 1
#define ACT_SIGMOID 2

// One wave computes one 16x16 tile of D = act(A @ W + bias).
// A: MxK row-major, W: KxN row-major. K%4==0, N%16==0. Ragged M handled by
// clamping loads (row m of D depends only on row m of A) and masking stores.
__global__ __launch_bounds__(32) void wmma_gemm_f32(
    const float* __restrict__ A, const float* __restrict__ W,
    const float* __restrict__ bias, float* __restrict__ Dst,
    int M, int K, int N, int act)
{
  const int tiles_n = N >> 4;
  const int tile = blockIdx.x;
  const int tm = tile / tiles_n;
  const int tn = tile - tm * tiles_n;
  const int m0 = tm << 4, n0 = tn << 4;
  const int lane = threadIdx.x & 31;
  const int half = lane >> 4;     // 0: K pair {0,1}, 1: K pair {2,3} (ISA layout)
  const int l16  = lane & 15;

  int arow = m0 + l16;
  if (arow >= M) arow = M - 1;    // clamp; masked at store
  const float* __restrict__ Arow = A + (size_t)arow * (size_t)K;
  const float* __restrict__ Wn   = W + (size_t)(n0 + l16);

  v8f acc = {0.f, 0.f, 0.f, 0.f, 0.f, 0.f, 0.f, 0.f};
  for (int k0 = 0; k0 < K; k0 += 4) {
    const int ka = k0 + (half << 1);
    v2f a, b;
    a.x = Arow[ka];
    a.y = Arow[ka + 1];
    b.x = Wn[(size_t)ka * (size_t)N];
    b.y = Wn[(size_t)(ka + 1) * (size_t)N];
    acc = __builtin_amdgcn_wmma_f32_16x16x4_f32(false, a, false, b,
                                                (short)0, acc, false, false);
  }

  const float bv = bias ? bias[n0 + l16] : 0.0f;
#pragma unroll
  for (int v = 0; v < 8; ++v) {
    const int row = m0 + half * 8 + v;   // C/D layout: VGPR v -> row v + 8*(lane/16)
    if (row < M) {
      float x = acc[v] + bv;
      if (act == ACT_RELU)         x = fmaxf(x, 0.0f);
      else if (act == ACT_SIGMOID) x = 1.0f / (1.0f + __expf(-x));
      Dst[(size_t)row * (size_t)N + (size_t)(n0 + l16)] = x;
    }
  }
}

// S[b, 0:100, :] = tmpSmi[b*100 ...], S[b, 100:145, :] = drug_gat[b]
__global__ void assemble_S(const float* __restrict__ tmpSmi,
                           const float* __restrict__ drug,
                           float* __restrict__ S)
{
  const int idx = blockIdx.x * blockDim.x + threadIdx.x;
  const int total = 8 * 145 * 64;
  if (idx >= total) return;
  const int d = idx & 63;
  const int r = idx >> 6;          // b*145 + i
  const int b = r / 145;
  const int i = r - b * 145;
  float v;
  if (i < 100) v = tmpSmi[((size_t)(b * 100 + i) << 6) + d];
  else         v = drug[((size_t)(b * 45 + (i - 100)) << 6) + d];
  S[idx] = v;
}

// Msmi[b,i,d] = (1/1000) * sum_j relu(S[b,i,d] + P[b,j,d])
// 5 S-rows per block (145 = 5*29 -> blocks never straddle a batch);
// each P load feeds 5 accumulators -> 5x less L2 read traffic.
#define SMI_RPB 5
__global__ void mean_relu_smi(const float* __restrict__ S,
                              const float* __restrict__ P,
                              float* __restrict__ Msmi)
{
  const int r0 = blockIdx.x * SMI_RPB;   // base row (b*145 + i)
  const int b  = r0 / 145;
  const int d  = threadIdx.x;            // 0..63
  float s[SMI_RPB], acc[SMI_RPB];
#pragma unroll
  for (int t = 0; t < SMI_RPB; ++t) {
    s[t] = S[((size_t)(r0 + t) << 6) + d];
    acc[t] = 0.0f;
  }
  const float* __restrict__ Pb = P + ((size_t)b * 1000 << 6) + d;
  for (int j = 0; j < 1000; ++j) {
    const float p = Pb[(size_t)j << 6];
#pragma unroll
    for (int t = 0; t < SMI_RPB; ++t)
      acc[t] += fmaxf(s[t] + p, 0.0f);
  }
#pragma unroll
  for (int t = 0; t < SMI_RPB; ++t)
    Msmi[((size_t)(r0 + t) << 6) + d] = acc[t] * (1.0f / 1000.0f);
}

// Mpro[b,j,d] = (1/145) * sum_i relu(P[b,j,d] + S[b,i,d])
// 8 P-rows per block (1000 % 8 == 0); each S load feeds 8 accumulators.
#define PRO_RPB 8
__global__ void mean_relu_pro(const float* __restrict__ S,
                              const float* __restrict__ P,
                              float* __restrict__ Mpro)
{
  const int r0 = blockIdx.x * PRO_RPB;   // base row (b*1000 + j)
  const int b  = r0 / 1000;
  const int d  = threadIdx.x;            // 0..63
  float p[PRO_RPB], acc[PRO_RPB];
#pragma unroll
  for (int t = 0; t < PRO_RPB; ++t) {
    p[t] = P[((size_t)(r0 + t) << 6) + d];
    acc[t] = 0.0f;
  }
  const float* __restrict__ Sb = S + ((size_t)b * 145 << 6) + d;
  for (int i = 0; i < 145; ++i) {
    const float sv = Sb[(size_t)i << 6];
#pragma unroll
    for (int t = 0; t < PRO_RPB; ++t)
      acc[t] += fmaxf(p[t] + sv, 0.0f);
  }
#pragma unroll
  for (int t = 0; t < PRO_RPB; ++t)
    Mpro[((size_t)(r0 + t) << 6) + d] = acc[t] * (1.0f / 145.0f);
}

// X (16 x 128): row b<8: [ mean_i S*(0.5+Gsmi) | mean_j pro_tf*(0.5+Gpro) ]
// rows 8..15 zero (padding for the 16-row WMMA MLP).
__global__ void pool_kernel(const float* __restrict__ S,
                            const float* __restrict__ Gsmi,
                            const float* __restrict__ pro_tf,
                            const float* __restrict__ Gpro,
                            float* __restrict__ X)
{
  const int b = blockIdx.x;        // 0..15
  const int d = threadIdx.x;       // 0..63
  if (b >= 8) {
    X[b * 128 + d] = 0.0f;
    X[b * 128 + 64 + d] = 0.0f;
    return;
  }
  float acc = 0.0f;
  for (int i = 0; i < 145; ++i) {
    const size_t o = (((size_t)b * 145 + i) << 6) + d;
    acc += S[o] * (0.5f + Gsmi[o]);
  }
  X[b * 128 + d] = acc * (1.0f / 145.0f);

  float accp = 0.0f;
  for (int j = 0; j < 1000; ++j) {
    const size_t o = (((size_t)b * 1000 + j) << 6) + d;
    accp += pro_tf[o] * (0.5f + Gpro[o]);
  }
  X[b * 128 + 64 + d] = accp * (1.0f / 1000.0f);
}

// out (8,2) = H3(8,512) @ w4(512,2) + b4
__global__ void final_kernel(const float* __restrict__ H3,
                             const float* __restrict__ w4,
                             const float* __restrict__ b4,
                             float* __restrict__ out)
{
  const int t = threadIdx.x;       // 0..15
  if (t >= 16) return;
  const int b = t >> 1;
  const int c = t & 1;
  float acc = b4[c];
  for (int k = 0; k < 512; ++k)
    acc += H3[b * 512 + k] * w4[k * 2 + c];
  out[b * 2 + c] = acc;
}

extern "C" void kernel_launch(void* const* d_in, const int* in_sizes, int n_in,
                              void* d_out, int out_size, void* d_ws, size_t ws_size,
                              hipStream_t stream)
{
  const float* smi_tf   = (const float*)d_in[0];
  const float* pro_tf   = (const float*)d_in[1];
  const float* drug_gat = (const float*)d_in[2];
  const float* w_att    = (const float*)d_in[3];
  const float* b_att    = (const float*)d_in[4];
  const float* w1       = (const float*)d_in[5];
  const float* b1       = (const float*)d_in[6];
  const float* w2       = (const float*)d_in[7];
  const float* b2       = (const float*)d_in[8];
  const float* w3       = (const float*)d_in[9];
  const float* b3       = (const float*)d_in[10];
  const float* w4       = (const float*)d_in[11];
  const float* b4       = (const float*)d_in[12];
  float* out = (float*)d_out;

  float* ws = (float*)d_ws;
  float* tmpSmi = ws; ws += 800 * 64;        // smi_att rows, contiguous (b*100+i)
  float* S      = ws; ws += 8 * 145 * 64;    // smi_attss
  float* P      = ws; ws += 8000 * 64;       // pro_att
  float* Msmi   = ws; ws += 8 * 145 * 64;
  float* Mpro   = ws; ws += 8000 * 64;
  float* Gsmi   = ws; ws += 8 * 145 * 64;
  float* Gpro   = ws; ws += 8000 * 64;
  float* X      = ws; ws += 16 * 128;
  float* H1     = ws; ws += 16 * 1024;
  float* H2     = ws; ws += 16 * 1024;
  float* H3     = ws; ws += 16 * 512;

  // 1) shared attention projections (WMMA)
  wmma_gemm_f32<<<50 * 4, 32, 0, stream>>>(smi_tf, w_att, b_att, tmpSmi, 800, 64, 64, ACT_NONE);
  wmma_gemm_f32<<<500 * 4, 32, 0, stream>>>(pro_tf, w_att, b_att, P, 8000, 64, 64, ACT_NONE);

  // 2) assemble smi_attss
  assemble_S<<<(8 * 145 * 64 + 255) / 256, 256, 0, stream>>>(tmpSmi, drug_gat, S);

  // 3) fused pairwise relu means (register-blocked)
  mean_relu_smi<<<(8 * 145) / SMI_RPB, 64, 0, stream>>>(S, P, Msmi);
  mean_relu_pro<<<(8 * 1000) / PRO_RPB, 64, 0, stream>>>(S, P, Mpro);

  // 4) sigmoid gates: sigmoid(mean @ w_att + b_att)   (WMMA, fused sigmoid)
  wmma_gemm_f32<<<73 * 4, 32, 0, stream>>>(Msmi, w_att, b_att, Gsmi, 1160, 64, 64, ACT_SIGMOID);
  wmma_gemm_f32<<<500 * 4, 32, 0, stream>>>(Mpro, w_att, b_att, Gpro, 8000, 64, 64, ACT_SIGMOID);

  // 5) gated pooling -> X (16 x 128, rows 8..15 zeroed)
  pool_kernel<<<16, 64, 0, stream>>>(S, Gsmi, pro_tf, Gpro, X);

  // 6) MLP head (WMMA for the three wide layers, scalar for 512->2)
  wmma_gemm_f32<<<1 * 64, 32, 0, stream>>>(X,  w1, b1, H1, 16, 128,  1024, ACT_RELU);
  wmma_gemm_f32<<<1 * 64, 32, 0, stream>>>(H1, w2, b2, H2, 16, 1024, 1024, ACT_RELU);
  wmma_gemm_f32<<<1 * 32, 32, 0, stream>>>(H2, w3, b3, H3, 16, 1024, 512,  ACT_RELU);
  final_kernel<<<1, 16, 0, stream>>>(H3, w4, b4, out);
}